// AttentionLayer_10170482557156
// MI455X (gfx1250) — compile-verified
//
#include <hip/hip_runtime.h>
#include <math.h>

// ---------------------------------------------------------------------------
// Multi-path GAT + semantic attention for MI455X (gfx1250, wave32, WMMA).
// K1 GEMM: workgroup-tiled (32x256), A/B panels staged in LDS via
//          GLOBAL_LOAD_ASYNC_TO_LDS_B128 (ASYNCcnt) when available,
//          compute on V_WMMA_F32_16X16X4_F32 (exact f32).
// Edge softmax/aggregation uses native f32 atomics, one wave per edge.
// ---------------------------------------------------------------------------

typedef float v2f __attribute__((ext_vector_type(2)));
typedef float v8f __attribute__((ext_vector_type(8)));
typedef int   v4i __attribute__((ext_vector_type(4)));

#define PATHS 3
#define DIN   256
#define DOUT  256
#define HEADS 4
#define CH    64
#define SHID  128

// LDS panel strides (floats): chosen for bank-conflict-free reads.
// B: lanes 16-31 read kk+2 -> 2*272 mod 64 = 32 -> disjoint bank halves.
// A: 16 rows * 20 mod 64 are all distinct banks; (kk,kk+1) pair = 1x b64.
#define BSTRIDE 272
#define ASTRIDE 20

#if defined(__HIP_DEVICE_COMPILE__) && \
    __has_builtin(__builtin_amdgcn_global_load_async_to_lds_b128) && \
    __has_builtin(__builtin_amdgcn_s_wait_asynccnt)
#define USE_ASYNC_LDS 1
#define AS1 __attribute__((address_space(1)))
#define AS3 __attribute__((address_space(3)))
#endif

__device__ __forceinline__ float leaky02(float v) { return v > 0.f ? v : 0.2f * v; }

__device__ __forceinline__ void atomicMaxF(float* addr, float v) {
    // Order-isomorphic int trick (no NaNs in this pipeline).
    if (v >= 0.f) atomicMax((int*)addr, __float_as_int(v));
    else          atomicMin((unsigned int*)addr, __float_as_uint(v));
}

// =============================== K1: x = h @ W[p] ===========================
// Workgroup tile: 32 rows x 256 cols (8 waves = 2 M-tiles x 4 N-tiles).
// Per 16-wide K chunk: stage A(32x16) + B(16x256) in LDS, then each wave runs
// 4 k-steps x 4 WMMA (f32 16x16x4) off LDS.
// A layout (16x4 f32): lane<16 -> K=k0,k0+1 ; lane>=16 -> K=k0+2,k0+3 (M=lane&15)
// B layout (4x16 f32): mirrored; C/D: VGPR r -> M = r + 8*(lane>=16), N=lane&15.
__global__ __launch_bounds__(256) void k_gemm_x(const float* __restrict__ h,
                                                const float* __restrict__ W,
                                                float* __restrict__ x,
                                                int nNodes) {
    __shared__ float lds[16 * BSTRIDE + 32 * ASTRIDE];
    float* ldsB = lds;
    float* ldsA = lds + 16 * BSTRIDE;

    const int mTiles  = nNodes >> 4;             // 3125
    const int mBlocks = (mTiles + 1) >> 1;       // 1563
    const int p       = blockIdx.x / mBlocks;
    const int mtBase  = blockIdx.x % mBlocks;

    const int tid  = threadIdx.x;
    const int wv   = tid >> 5;
    const int mtl  = wv >> 2;                    // 0..1 (M sub-tile)
    const int nt   = wv & 3;                     // 0..3 (N sub-tile, 64 wide)
    const int lane = tid & 31;
    const int l16  = lane & 15;
    const int half = lane >> 4;

    const float* Wp = W + (size_t)p * DIN * DOUT;

    // Staging maps.
    const int bKK = tid >> 4;                    // 0..15 (B panel row)
    const int bC0 = (tid & 15) * 16;             // B col segment (16 floats)
    const int aRowL = tid >> 2;                  // 0..31 for tid<128
    const int aK4   = (tid & 3) * 4;             // 4-float segment in row
    int aRowG = mtBase * 32 + aRowL;
    if (aRowG > nNodes - 1) aRowG = nNodes - 1;  // clamp for ragged last block

    v8f acc0 = {}, acc1 = {}, acc2 = {}, acc3 = {};

    for (int k0 = 0; k0 < DIN; k0 += 16) {
        __syncthreads();   // previous chunk's LDS reads complete
        {
            const float* gB = Wp + (size_t)(k0 + bKK) * DOUT + bC0;
            float*       lB = ldsB + bKK * BSTRIDE + bC0;
#ifdef USE_ASYNC_LDS
#pragma unroll
            for (int i = 0; i < 4; ++i)
                __builtin_amdgcn_global_load_async_to_lds_b128(
                    (AS1 v4i*)(gB + 4 * i), (AS3 v4i*)(lB + 4 * i), 0, 0);
#else
#pragma unroll
            for (int i = 0; i < 4; ++i)
                *(float4*)(lB + 4 * i) = *(const float4*)(gB + 4 * i);
#endif
        }
        if (tid < 128) {
            const float* gA = h + (size_t)aRowG * DIN + k0 + aK4;
            float*       lA = ldsA + aRowL * ASTRIDE + aK4;
#ifdef USE_ASYNC_LDS
            __builtin_amdgcn_global_load_async_to_lds_b128(
                (AS1 v4i*)gA, (AS3 v4i*)lA, 0, 0);
#else
            *(float4*)lA = *(const float4*)gA;
#endif
        }
#ifdef USE_ASYNC_LDS
        __builtin_amdgcn_s_wait_asynccnt(0);
#endif
        __syncthreads();   // panels visible to all waves

#pragma unroll
        for (int kl = 0; kl < 16; kl += 4) {
            const int kk = kl + half * 2;
            v2f a = *(const v2f*)(ldsA + (mtl * 16 + l16) * ASTRIDE + kk);
            const float* b0p = ldsB + kk * BSTRIDE + nt * 64 + l16;
            const float* b1p = b0p + BSTRIDE;
            v2f b0 = { b0p[0],  b1p[0]  };
            v2f b1 = { b0p[16], b1p[16] };
            v2f b2 = { b0p[32], b1p[32] };
            v2f b3 = { b0p[48], b1p[48] };
            acc0 = __builtin_amdgcn_wmma_f32_16x16x4_f32(false, a, false, b0, (short)0, acc0, false, false);
            acc1 = __builtin_amdgcn_wmma_f32_16x16x4_f32(false, a, false, b1, (short)0, acc1, false, false);
            acc2 = __builtin_amdgcn_wmma_f32_16x16x4_f32(false, a, false, b2, (short)0, acc2, false, false);
            acc3 = __builtin_amdgcn_wmma_f32_16x16x4_f32(false, a, false, b3, (short)0, acc3, false, false);
        }
    }

    const int mt = mtBase * 2 + mtl;
    if (mt < mTiles) {
        float* xOut = x + (size_t)p * nNodes * DOUT;
#pragma unroll
        for (int r = 0; r < 8; ++r) {
            const int node = mt * 16 + r + half * 8;
            float* dst = xOut + (size_t)node * DOUT + nt * 64 + l16;
            dst[0]  = acc0[r];
            dst[16] = acc1[r];
            dst[32] = acc2[r];
            dst[48] = acc3[r];
        }
    }
}

// ====================== K2: per-node scores + m init (self-loop) ============
__global__ __launch_bounds__(256) void k_scores(const float* __restrict__ x,
                                                const float* __restrict__ att_src,
                                                const float* __restrict__ att_dst,
                                                float* __restrict__ aSrc,
                                                float* __restrict__ aDst,
                                                float* __restrict__ mOut,
                                                int nNodes) {
    const int idx   = blockIdx.x * blockDim.x + threadIdx.x;
    const int total = PATHS * nNodes * HEADS;
    if (idx >= total) return;
    const int hd = idx & (HEADS - 1);
    const int n  = (idx >> 2) % nNodes;
    const int p  = idx / (nNodes * HEADS);

    const float* xr = x + ((size_t)p * nNodes + n) * DOUT + hd * CH;
    const float* as = att_src + (p * HEADS + hd) * CH;
    const float* ad = att_dst + (p * HEADS + hd) * CH;
    float ssum = 0.f, dsum = 0.f;
#pragma unroll 4
    for (int c = 0; c < CH; c += 4) {
        float4 xv = *(const float4*)(xr + c);
        float4 av = *(const float4*)(as + c);
        float4 dv = *(const float4*)(ad + c);
        ssum += xv.x * av.x + xv.y * av.y + xv.z * av.z + xv.w * av.w;
        dsum += xv.x * dv.x + xv.y * dv.y + xv.z * dv.z + xv.w * dv.w;
    }
    aSrc[idx] = ssum;
    aDst[idx] = dsum;
    mOut[idx] = leaky02(ssum + dsum);   // self-loop seeds the segment max
}

// ========================= K3: segment max over edges =======================
__global__ __launch_bounds__(256) void k_segmax(const int* __restrict__ ei,
                                                const float* __restrict__ aSrc,
                                                const float* __restrict__ aDst,
                                                float* __restrict__ m,
                                                int nNodes, int nE) {
    const int idx = blockIdx.x * blockDim.x + threadIdx.x;
    if (idx >= PATHS * nE) return;
    const int e = idx % nE;
    const int p = idx / nE;
    const int src = ei[(size_t)p * 2 * nE + e];
    const int dst = ei[(size_t)p * 2 * nE + nE + e];
    const float* as = aSrc + ((size_t)p * nNodes + src) * HEADS;
    const float* ad = aDst + ((size_t)p * nNodes + dst) * HEADS;
    float* mrow = m + ((size_t)p * nNodes + dst) * HEADS;
#pragma unroll
    for (int hd = 0; hd < HEADS; ++hd)
        atomicMaxF(&mrow[hd], leaky02(as[hd] + ad[hd]));
}

// ================= K4a: s init with self-loop contribution ==================
__global__ __launch_bounds__(256) void k_sinit(const float* __restrict__ aSrc,
                                               const float* __restrict__ aDst,
                                               const float* __restrict__ m,
                                               float* __restrict__ s,
                                               int nNodes) {
    const int idx = blockIdx.x * blockDim.x + threadIdx.x;
    if (idx >= PATHS * nNodes * HEADS) return;
    const float e = leaky02(aSrc[idx] + aDst[idx]);
    s[idx] = __expf(e - m[idx]);
}

// ===================== K4b: segment sum of exp over edges ===================
__global__ __launch_bounds__(256) void k_segsum(const int* __restrict__ ei,
                                                const float* __restrict__ aSrc,
                                                const float* __restrict__ aDst,
                                                const float* __restrict__ m,
                                                float* __restrict__ s,
                                                int nNodes, int nE) {
    const int idx = blockIdx.x * blockDim.x + threadIdx.x;
    if (idx >= PATHS * nE) return;
    const int e = idx % nE;
    const int p = idx / nE;
    const int src = ei[(size_t)p * 2 * nE + e];
    const int dst = ei[(size_t)p * 2 * nE + nE + e];
    const float* as = aSrc + ((size_t)p * nNodes + src) * HEADS;
    const float* ad = aDst + ((size_t)p * nNodes + dst) * HEADS;
    const float* mrow = m + ((size_t)p * nNodes + dst) * HEADS;
    float* srow = s + ((size_t)p * nNodes + dst) * HEADS;
#pragma unroll
    for (int hd = 0; hd < HEADS; ++hd) {
        const float ev = leaky02(as[hd] + ad[hd]);
        unsafeAtomicAdd(&srow[hd], __expf(ev - mrow[hd]));
    }
}

// ========= K5a: z init = bias + self-loop alpha * x (per 4 channels) ========
__global__ __launch_bounds__(256) void k_zinit(const float* __restrict__ x,
                                               const float* __restrict__ bias,
                                               const float* __restrict__ aSrc,
                                               const float* __restrict__ aDst,
                                               const float* __restrict__ m,
                                               const float* __restrict__ s,
                                               float* __restrict__ z,
                                               int nNodes) {
    const int idx = blockIdx.x * blockDim.x + threadIdx.x;   // (p, n, d/4)
    if (idx >= PATHS * nNodes * (DOUT / 4)) return;
    const int d = (idx & 63) * 4;
    const int n = (idx >> 6) % nNodes;
    const int p = idx / (nNodes * 64);
    const int hd = d >> 6;

    const size_t si = ((size_t)p * nNodes + n) * HEADS + hd;
    const float e = leaky02(aSrc[si] + aDst[si]);
    const float alpha = __expf(e - m[si]) / (s[si] + 1e-16f);

    float4 xv = *(const float4*)(x + ((size_t)p * nNodes + n) * DOUT + d);
    float4 bv = *(const float4*)(bias + p * DOUT + d);
    float4 o;
    o.x = bv.x + alpha * xv.x;  o.y = bv.y + alpha * xv.y;
    o.z = bv.z + alpha * xv.z;  o.w = bv.w + alpha * xv.w;
    *(float4*)(z + ((size_t)n * PATHS + p) * DOUT + d) = o;
}

// ============ K5b: edge aggregation, one wave per (p, edge) =================
__global__ __launch_bounds__(256) void k_aggregate(const int* __restrict__ ei,
                                                   const float* __restrict__ x,
                                                   const float* __restrict__ aSrc,
                                                   const float* __restrict__ aDst,
                                                   const float* __restrict__ m,
                                                   const float* __restrict__ s,
                                                   float* __restrict__ z,
                                                   int nNodes, int nE) {
    const int wave = blockIdx.x * (blockDim.x >> 5) + (threadIdx.x >> 5);
    if (wave >= PATHS * nE) return;
    const int e = wave % nE;
    const int p = wave / nE;
    const int lane = threadIdx.x & 31;
    const int src = ei[(size_t)p * 2 * nE + e];
    const int dst = ei[(size_t)p * 2 * nE + nE + e];

    const int d  = lane * 8;         // 8 floats per lane, all in one head
    const int hd = d >> 6;
    const size_t srcI = ((size_t)p * nNodes + src) * HEADS + hd;
    const size_t dstI = ((size_t)p * nNodes + dst) * HEADS + hd;
    const float ev = leaky02(aSrc[srcI] + aDst[dstI]);
    const float alpha = __expf(ev - m[dstI]) / (s[dstI] + 1e-16f);

    const float* xr = x + ((size_t)p * nNodes + src) * DOUT + d;
    float* zr = z + ((size_t)dst * PATHS + p) * DOUT + d;
    float4 x0 = *(const float4*)(xr);
    float4 x1 = *(const float4*)(xr + 4);
    unsafeAtomicAdd(zr + 0, alpha * x0.x);
    unsafeAtomicAdd(zr + 1, alpha * x0.y);
    unsafeAtomicAdd(zr + 2, alpha * x0.z);
    unsafeAtomicAdd(zr + 3, alpha * x0.w);
    unsafeAtomicAdd(zr + 4, alpha * x1.x);
    unsafeAtomicAdd(zr + 5, alpha * x1.y);
    unsafeAtomicAdd(zr + 6, alpha * x1.z);
    unsafeAtomicAdd(zr + 7, alpha * x1.w);
}

// ============== K6a: t = tanh(z2d @ sW1 + sb1), WMMA f32 ====================
// z2d: [N*P, 256] row-major, sW1: [256,128] (L2-resident). Wave tile 16x64.
__global__ __launch_bounds__(256) void k_gemm_t(const float* __restrict__ z,
                                                const float* __restrict__ sW1,
                                                const float* __restrict__ sb1,
                                                float* __restrict__ t,
                                                int rows) {
    const int wave   = blockIdx.x * (blockDim.x >> 5) + (threadIdx.x >> 5);
    const int mTiles = rows >> 4;                 // 9375
    if (wave >= mTiles * 2) return;

    const int nt   = wave & 1;
    const int mt   = wave >> 1;
    const int lane = threadIdx.x & 31;
    const int l16  = lane & 15;
    const int half = lane >> 4;

    const float* zRow = z + (size_t)(mt * 16 + l16) * DOUT;
    const int colBase = nt * 64 + l16;

    v8f acc0 = {}, acc1 = {}, acc2 = {}, acc3 = {};
#pragma unroll 4
    for (int k0 = 0; k0 < DOUT; k0 += 4) {
        const int kk = k0 + half * 2;
        v2f a = *(const v2f*)(zRow + kk);
        const float* wk0 = sW1 + (size_t)kk * SHID + colBase;
        const float* wk1 = wk0 + SHID;
        v2f b0 = { wk0[0],  wk1[0]  };
        v2f b1 = { wk0[16], wk1[16] };
        v2f b2 = { wk0[32], wk1[32] };
        v2f b3 = { wk0[48], wk1[48] };
        acc0 = __builtin_amdgcn_wmma_f32_16x16x4_f32(false, a, false, b0, (short)0, acc0, false, false);
        acc1 = __builtin_amdgcn_wmma_f32_16x16x4_f32(false, a, false, b1, (short)0, acc1, false, false);
        acc2 = __builtin_amdgcn_wmma_f32_16x16x4_f32(false, a, false, b2, (short)0, acc2, false, false);
        acc3 = __builtin_amdgcn_wmma_f32_16x16x4_f32(false, a, false, b3, (short)0, acc3, false, false);
    }

    const float bb0 = sb1[colBase +  0];
    const float bb1 = sb1[colBase + 16];
    const float bb2 = sb1[colBase + 32];
    const float bb3 = sb1[colBase + 48];
#pragma unroll
    for (int r = 0; r < 8; ++r) {
        const int row = mt * 16 + r + half * 8;
        float* dst = t + (size_t)row * SHID + colBase;
        dst[0]  = tanhf(acc0[r] + bb0);
        dst[16] = tanhf(acc1[r] + bb1);
        dst[32] = tanhf(acc2[r] + bb2);
        dst[48] = tanhf(acc3[r] + bb3);
    }
}

// ================= K6b: w[row] = dot(t[row,:], sW2) =========================
__global__ __launch_bounds__(256) void k_wscore(const float* __restrict__ t,
                                                const float* __restrict__ sW2,
                                                float* __restrict__ w,
                                                int rows) {
    const int r = blockIdx.x * blockDim.x + threadIdx.x;
    if (r >= rows) return;
    const float* tr = t + (size_t)r * SHID;
    float sum = 0.f;
#pragma unroll 8
    for (int j = 0; j < SHID; j += 4) {
        float4 tv = *(const float4*)(tr + j);
        float4 wv = *(const float4*)(sW2 + j);
        sum += tv.x * wv.x + tv.y * wv.y + tv.z * wv.z + tv.w * wv.w;
    }
    w[r] = sum;
}

// ============= K6c: beta = softmax_P(w), out = sum_p beta_p z_p =============
__global__ __launch_bounds__(256) void k_final(const float* __restrict__ z,
                                               const float* __restrict__ w,
                                               float* __restrict__ out,
                                               int nNodes) {
    const int idx = blockIdx.x * blockDim.x + threadIdx.x;   // (n, d/4)
    if (idx >= nNodes * (DOUT / 4)) return;
    const int d = (idx & 63) * 4;
    const int n = idx >> 6;

    const float w0 = w[n * PATHS + 0];
    const float w1 = w[n * PATHS + 1];
    const float w2 = w[n * PATHS + 2];
    const float mx = fmaxf(w0, fmaxf(w1, w2));
    float b0 = __expf(w0 - mx), b1 = __expf(w1 - mx), b2 = __expf(w2 - mx);
    const float inv = 1.f / (b0 + b1 + b2);
    b0 *= inv; b1 *= inv; b2 *= inv;

    const float* zr = z + (size_t)n * PATHS * DOUT + d;
    float4 z0 = *(const float4*)(zr);
    float4 z1 = *(const float4*)(zr + DOUT);
    float4 z2 = *(const float4*)(zr + 2 * DOUT);
    float4 o;
    o.x = b0 * z0.x + b1 * z1.x + b2 * z2.x;
    o.y = b0 * z0.y + b1 * z1.y + b2 * z2.y;
    o.z = b0 * z0.z + b1 * z1.z + b2 * z2.z;
    o.w = b0 * z0.w + b1 * z1.w + b2 * z2.w;
    *(float4*)(out + (size_t)n * DOUT + d) = o;
}

// ===========================================================================
extern "C" void kernel_launch(void* const* d_in, const int* in_sizes, int n_in,
                              void* d_out, int out_size, void* d_ws, size_t ws_size,
                              hipStream_t stream) {
    const float* h       = (const float*)d_in[0];
    const int*   ei      = (const int*)d_in[1];
    const float* W       = (const float*)d_in[2];
    const float* att_src = (const float*)d_in[3];
    const float* att_dst = (const float*)d_in[4];
    const float* bias    = (const float*)d_in[5];
    const float* sW1     = (const float*)d_in[6];
    const float* sb1     = (const float*)d_in[7];
    const float* sW2     = (const float*)d_in[8];
    float* out = (float*)d_out;

    const int nNodes = in_sizes[0] / DIN;          // 50000
    const int nE     = in_sizes[1] / (PATHS * 2);  // 400000

    // Workspace layout (all regions fully rewritten every call):
    char* ws = (char*)d_ws;
    const size_t xBytes = (size_t)PATHS * nNodes * DOUT * sizeof(float);   // 153.6 MB
    const size_t aBytes = (size_t)PATHS * nNodes * HEADS * sizeof(float);  // 2.4 MB
    float* x    = (float*)ws;
    float* z    = (float*)(ws + xBytes);
    char*  ws2  = ws + 2 * xBytes;
    float* aSrc = (float*)(ws2);
    float* aDst = (float*)(ws2 + aBytes);
    float* m    = (float*)(ws2 + 2 * aBytes);
    float* s    = (float*)(ws2 + 3 * aBytes);
    float* wsc  = (float*)(ws2 + 4 * aBytes);
    float* t    = x;   // x is dead after K5b; t (76.8 MB) reuses its region

    const int mTiles  = nNodes / 16;
    const int mBlocks = (mTiles + 1) / 2;
    k_gemm_x<<<PATHS * mBlocks, 256, 0, stream>>>(h, W, x, nNodes);

    const int totPNH = PATHS * nNodes * HEADS;
    k_scores<<<(totPNH + 255) / 256, 256, 0, stream>>>(x, att_src, att_dst, aSrc, aDst, m, nNodes);

    const int totPE = PATHS * nE;
    k_segmax<<<(totPE + 255) / 256, 256, 0, stream>>>(ei, aSrc, aDst, m, nNodes, nE);
    k_sinit<<<(totPNH + 255) / 256, 256, 0, stream>>>(aSrc, aDst, m, s, nNodes);
    k_segsum<<<(totPE + 255) / 256, 256, 0, stream>>>(ei, aSrc, aDst, m, s, nNodes, nE);

    const int totZ4 = PATHS * nNodes * (DOUT / 4);
    k_zinit<<<(totZ4 + 255) / 256, 256, 0, stream>>>(x, bias, aSrc, aDst, m, s, z, nNodes);
    k_aggregate<<<(totPE + 7) / 8, 256, 0, stream>>>(ei, x, aSrc, aDst, m, s, z, nNodes, nE);

    const int rows   = nNodes * PATHS;
    const int waves6 = (rows / 16) * 2;
    k_gemm_t<<<(waves6 + 7) / 8, 256, 0, stream>>>(z, sW1, sb1, t, rows);
    k_wscore<<<(rows + 255) / 256, 256, 0, stream>>>(t, sW2, wsc, rows);

    const int totO4 = nNodes * (DOUT / 4);
    k_final<<<(totO4 + 255) / 256, 256, 0, stream>>>(z, wsc, out, nNodes);
}